// MaskcomputeMoE3_56547539419491
// MI455X (gfx1250) — compile-verified
//
#include <hip/hip_runtime.h>
#include <hip/hip_bf16.h>
#include <stdint.h>

typedef __attribute__((ext_vector_type(2))) float v2f;
typedef __attribute__((ext_vector_type(4))) float v4f;
typedef __attribute__((ext_vector_type(8))) float v8f;

#define C_DIM 768
#define L_DIM 512
#define B_DIM 8
#define E_DIM 32

// ---------------------------------------------------------------------------
// Kernel 0: pos[l][c] (RoPE-ones), batch independent.
// grid = dim3(3, 512), block = 256  (3*256 = 768 columns)
// ---------------------------------------------------------------------------
__global__ void pos_kernel(float* __restrict__ pos) {
  const int l = blockIdx.y;
  const int c = blockIdx.x * blockDim.x + threadIdx.x;
  const int j = (c < 384) ? c : (c - 384);
  // theta = 12000^(-j/384) = exp(-j * ln(12000)/384)
  const float theta = __expf(-(float)j * 0.024460057106171191f);
  const float f = (float)l * theta;
  float s, cs;
  __sincosf(f, &s, &cs);
  pos[l * C_DIM + c] = (c < 384) ? (cs - s) : (s + cs);
}

// ---------------------------------------------------------------------------
// Kernel 1: h = gelu(pos @ W1 + b1), 512x768 = 32 Mtiles x 48 Ntiles,
// one wave per 16x16 tile via V_WMMA_F32_16X16X4_F32, 192 K-steps.
// grid = 192 blocks x 256 threads (8 waves/block)
// ---------------------------------------------------------------------------
__global__ void gemm_h_kernel(const float* __restrict__ pos,
                              const float* __restrict__ W1,
                              const float* __restrict__ b1,
                              float* __restrict__ h) {
  const int wave = (blockIdx.x * blockDim.x + threadIdx.x) >> 5;
  const int lane = threadIdx.x & 31;
  const int mt = wave / 48;           // 0..31
  const int nt = wave - mt * 48;      // 0..47
  const int lo = lane & 15;
  const int hi = lane >> 4;
  const int mrow = mt * 16 + lo;
  const int ncol = nt * 16 + lo;

  v8f acc = {};
  const float* aptr = pos + mrow * C_DIM + 2 * hi;      // A[m][k0+2hi], [.. +1]
  const float* bptr = W1 + (2 * hi) * C_DIM + ncol;     // B[k0+2hi][n], [k0+2hi+1][n]
  for (int k0 = 0; k0 < C_DIM; k0 += 4) {
    v2f a; a.x = aptr[0];     a.y = aptr[1];
    v2f b; b.x = bptr[0];     b.y = bptr[C_DIM];
    acc = __builtin_amdgcn_wmma_f32_16x16x4_f32(false, a, false, b,
                                                (short)0, acc, false, false);
    aptr += 4;
    bptr += 4 * C_DIM;
  }
  const float bias = b1[ncol];
#pragma unroll
  for (int r = 0; r < 8; ++r) {
    const float x = acc[r] + bias;
    const float gl = 0.5f * x * (1.0f + erff(x * 0.70710678118654752f)); // exact gelu
    h[(mt * 16 + r + 8 * hi) * C_DIM + ncol] = gl;
  }
}

// ---------------------------------------------------------------------------
// Kernel 2: logits = h @ W2[:, :32]  -> (512 x 32). 32 Mtiles x 2 Ntiles = 64 waves.
// grid = 8 blocks x 256 threads
// ---------------------------------------------------------------------------
__global__ void gemm_logits_kernel(const float* __restrict__ h,
                                   const float* __restrict__ W2,
                                   float* __restrict__ logits) {
  const int wave = (blockIdx.x * blockDim.x + threadIdx.x) >> 5;
  const int lane = threadIdx.x & 31;
  const int mt = wave >> 1;           // 0..31
  const int nt = wave & 1;            // 0..1
  const int lo = lane & 15;
  const int hi = lane >> 4;
  const int mrow = mt * 16 + lo;
  const int ncol = nt * 16 + lo;      // 0..31

  v8f acc = {};
  const float* aptr = h + mrow * C_DIM + 2 * hi;
  const float* bptr = W2 + (2 * hi) * (2 * E_DIM) + ncol;   // W2 row stride = 64
  for (int k0 = 0; k0 < C_DIM; k0 += 4) {
    v2f a; a.x = aptr[0]; a.y = aptr[1];
    v2f b; b.x = bptr[0]; b.y = bptr[2 * E_DIM];
    acc = __builtin_amdgcn_wmma_f32_16x16x4_f32(false, a, false, b,
                                                (short)0, acc, false, false);
    aptr += 4;
    bptr += 4 * (2 * E_DIM);
  }
#pragma unroll
  for (int r = 0; r < 8; ++r)
    logits[(mt * 16 + r + 8 * hi) * E_DIM + ncol] = acc[r];
}

// ---------------------------------------------------------------------------
// JAX threefry2x32, key = (0, 42)  [jax.random.key(42)]
// ---------------------------------------------------------------------------
__device__ __forceinline__ uint32_t rotl32(uint32_t x, int r) {
  return (x << r) | (x >> (32 - r));
}
__device__ __forceinline__ void threefry2x32_42(uint32_t x0, uint32_t x1,
                                                uint32_t* o0, uint32_t* o1) {
  const uint32_t k0 = 0u, k1 = 42u;
  const uint32_t k2 = k0 ^ k1 ^ 0x1BD11BDAu;
  x0 += k0; x1 += k1;
#define TF_R4(a, b, c, d) \
  x0 += x1; x1 = rotl32(x1, a); x1 ^= x0; \
  x0 += x1; x1 = rotl32(x1, b); x1 ^= x0; \
  x0 += x1; x1 = rotl32(x1, c); x1 ^= x0; \
  x0 += x1; x1 = rotl32(x1, d); x1 ^= x0;
  TF_R4(13, 15, 26, 6)  x0 += k1; x1 += k2 + 1u;
  TF_R4(17, 29, 16, 24) x0 += k2; x1 += k0 + 2u;
  TF_R4(13, 15, 26, 6)  x0 += k0; x1 += k1 + 3u;
  TF_R4(17, 29, 16, 24) x0 += k1; x1 += k2 + 4u;
  TF_R4(13, 15, 26, 6)  x0 += k2; x1 += k0 + 5u;
#undef TF_R4
  *o0 = x0; *o1 = x1;
}

// ---------------------------------------------------------------------------
// Kernel 3: per-(b,l) wave: lane e computes logits[l,e] + gumbel(b,l,e),
// wave32 butterfly argmax (first index wins ties). grid = 512 x 256.
// ---------------------------------------------------------------------------
__global__ void gumbel_argmax_kernel(const float* __restrict__ logits,
                                     int* __restrict__ winner) {
  const int gid = blockIdx.x * blockDim.x + threadIdx.x;     // == flat (b,l,e)
  const int e = gid & 31;
  const int wid = gid >> 5;                                  // b*512 + l
  const int l = wid & (L_DIM - 1);

  const uint32_t i = (uint32_t)gid;
  const uint32_t half = (uint32_t)(B_DIM * L_DIM * E_DIM) / 2u;  // 65536
  uint32_t o0, o1, bits;
  if (i < half) { threefry2x32_42(i, i + half, &o0, &o1); bits = o0; }
  else          { threefry2x32_42(i - half, i, &o0, &o1); bits = o1; }

  const float f = __uint_as_float((bits >> 9) | 0x3f800000u) - 1.0f;
  float u = f * (1.0f - 1e-8f) + 1e-8f;
  u = fmaxf(1e-8f, u);
  const float g = -logf(-logf(u));

  float val = logits[l * E_DIM + e] + g;   // tau>0 scaling preserves argmax
  int idx = e;
#pragma unroll
  for (int off = 16; off > 0; off >>= 1) {
    const float v2 = __shfl_xor(val, off, 32);
    const int   i2 = __shfl_xor(idx, off, 32);
    if (v2 > val || (v2 == val && i2 < idx)) { val = v2; idx = i2; }
  }
  if (e == 0) winner[wid] = idx;
}

// ---------------------------------------------------------------------------
// Kernel 4: per-(b,e) winner count over L; top-100 threshold degenerates to:
// count >= 100 -> only winners survive; else everything survives.
// 1 block x 256 threads, no atomics (deterministic, no init needed).
// ---------------------------------------------------------------------------
__global__ void count_kernel(const int* __restrict__ winner,
                             int* __restrict__ keepAll) {
  const int t = threadIdx.x;          // 0..255
  const int b = t >> 5, e = t & 31;
  int cnt = 0;
  for (int l = 0; l < L_DIM; ++l)
    cnt += (winner[b * L_DIM + l] == e) ? 1 : 0;
  keepAll[t] = (cnt < 100) ? 1 : 0;
}

// ---------------------------------------------------------------------------
// Kernel 5: expert_outputs[e,b,l,c] = P[b,e,l] * input[b,l,c].
// THE bandwidth kernel: 402 MB nontemporal b128 stores, cached b128 loads.
// grid = dim3(L=512, B=8, E=32), block = 192 (one float4 per thread per row).
// ---------------------------------------------------------------------------
__global__ void expert_out_kernel(const float* __restrict__ x,
                                  const int* __restrict__ winner,
                                  const int* __restrict__ keepAll,
                                  float* __restrict__ out) {
  const int l = blockIdx.x, b = blockIdx.y, e = blockIdx.z;
  const int c4 = threadIdx.x;                        // 0..191
  const int w = winner[b * L_DIM + l];
  float s;
  if (w == e) s = 1.0001f;
  else        s = keepAll[b * E_DIM + e] ? 1e-4f : 0.0f;

  const v4f* __restrict__ xin = (const v4f*)(x + ((size_t)(b * L_DIM + l)) * C_DIM);
  v4f v = xin[c4];
  v *= s;
  const size_t t = (((size_t)(e * B_DIM + b) * L_DIM + l) * (C_DIM / 4)) + c4;
  __builtin_nontemporal_store(v, ((v4f*)out) + t);
}

// ---------------------------------------------------------------------------
// Kernel 6: P[b,e,l]. grid = 512 x 256 over B*E*L = 131072.
// ---------------------------------------------------------------------------
__global__ void p_kernel(const int* __restrict__ winner,
                         const int* __restrict__ keepAll,
                         float* __restrict__ P) {
  const int id = blockIdx.x * blockDim.x + threadIdx.x;  // b*(E*L) + e*L + l
  const int l = id & (L_DIM - 1);
  const int e = (id >> 9) & (E_DIM - 1);
  const int b = id >> 14;
  const int w = winner[b * L_DIM + l];
  float s;
  if (w == e) s = 1.0001f;
  else        s = keepAll[b * E_DIM + e] ? 1e-4f : 0.0f;
  P[id] = s;
}

// ---------------------------------------------------------------------------
// Launch. Inputs: [input_features(B,L,C) f32, W1(C,C) f32, b1(C) f32, W2(C,2E) f32]
// Output: expert_outputs (E*B,L,C) f32 then P (B,E,L) f32, concatenated.
// ---------------------------------------------------------------------------
extern "C" void kernel_launch(void* const* d_in, const int* in_sizes, int n_in,
                              void* d_out, int out_size, void* d_ws, size_t ws_size,
                              hipStream_t stream) {
  const float* x  = (const float*)d_in[0];
  const float* W1 = (const float*)d_in[1];
  const float* b1 = (const float*)d_in[2];
  const float* W2 = (const float*)d_in[3];

  float* ws_f = (float*)d_ws;
  float* pos    = ws_f;                       // 512*768
  float* h      = ws_f + 393216;              // 512*768
  float* logits = ws_f + 786432;              // 512*32
  int*   winner  = (int*)(ws_f + 802816);     // 4096
  int*   keepAll = (int*)(ws_f + 806912);     // 256

  float* expert_out = (float*)d_out;                                   // E*B*L*C
  float* P          = (float*)d_out + (size_t)E_DIM * B_DIM * L_DIM * C_DIM;

  pos_kernel<<<dim3(3, L_DIM), 256, 0, stream>>>(pos);
  gemm_h_kernel<<<192, 256, 0, stream>>>(pos, W1, b1, h);          // 1536 tiles
  gemm_logits_kernel<<<8, 256, 0, stream>>>(h, W2, logits);        // 64 tiles
  gumbel_argmax_kernel<<<512, 256, 0, stream>>>(logits, winner);   // B*L waves
  count_kernel<<<1, 256, 0, stream>>>(winner, keepAll);
  expert_out_kernel<<<dim3(L_DIM, B_DIM, E_DIM), C_DIM / 4, 0, stream>>>(
      x, winner, keepAll, expert_out);
  p_kernel<<<512, 256, 0, stream>>>(winner, keepAll, P);
}